// EdgeConv_11373073400090
// MI455X (gfx1250) — compile-verified
//
#include <hip/hip_runtime.h>

typedef __attribute__((ext_vector_type(2))) float v2f;
typedef __attribute__((ext_vector_type(8))) float v8f;

// Problem constants (from reference): B=4, N=32768, K=20, C=64, OUT=64
constexpr int Bn = 4;
constexpr int Nn = 32768;
constexpr int Kn = 20;
constexpr int Cn = 64;
constexpr int On = 64;

constexpr int WAVES   = 4;      // waves per block (128 threads)
constexpr int A_STRIDE = 132;   // 16x128 A-tile row stride (floats), padded: bank-conflict-free, 16B-aligned rows

__global__ __launch_bounds__(WAVES * 32)
void EdgeConv_wmma_kernel(const float* __restrict__ x,
                          const int*   __restrict__ adj,
                          const float* __restrict__ W,
                          const float* __restrict__ bias,
                          float*       __restrict__ out)
{
    // Per-wave LDS: 20 neighbor rows staged contiguously (the "flat" gather array),
    // plus the 16x128 f32 A matrix (mean edge features) for WMMA.
    __shared__ __align__(16) float s_concat[WAVES][Kn * Cn];      // 1280 floats / wave
    __shared__ __align__(16) float s_A[WAVES][16 * A_STRIDE];     // 2112 floats / wave

    const int lane = threadIdx.x & 31;
    const int wid  = threadIdx.x >> 5;
    const int wave_global = blockIdx.x * WAVES + wid;
    const int base_point  = wave_global * 16;          // 16 points per wave

    float* concat = s_concat[wid];
    float* Atile  = s_A[wid];

    const int hi  = lane >> 4;   // 0: lanes 0-15, 1: lanes 16-31
    const int l15 = lane & 15;

    // ---------------- Phase 1: build A = mean edge features (16 x 128) ----------------
    // e_mean[c]      = (1/20) * sum_{j=20c..20c+19} concat[j]  -  x[n][c]   (c < 64)
    // e_mean[64 + c] = x[n][c]
    for (int p = 0; p < 16; ++p) {
        const int point = base_point + p;
        const int b = point >> 15;            // / N  (N = 32768)
        const int n = point & (Nn - 1);
        const float* xb = x + (size_t)b * Nn * Cn;

        // neighbor indices, one per lane for lanes 0..19
        int vidx = 0;
        if (lane < Kn) vidx = adj[(size_t)point * Kn + lane];

        // Stage the 20 neighbor rows (coalesced: 2 floats per lane per row)
        for (int k = 0; k < Kn; ++k) {
            const int row = __shfl(vidx, k, 32);
            const float2 v = *(const float2*)(xb + (size_t)row * Cn + lane * 2);
            *(float2*)&concat[k * Cn + lane * 2] = v;
        }

        // Center-point features for this point
        const float xc0 = xb[(size_t)n * Cn + lane];        // c = lane
        const float xc1 = xb[(size_t)n * Cn + lane + 32];   // c = lane + 32

        // 20-wide window sums over the flat gather array (same-wave LDS: HW keeps in order)
        float s0 = 0.f, s1 = 0.f;
        #pragma unroll
        for (int t = 0; t < Kn; ++t) {
            s0 += concat[Kn * lane        + t];
            s1 += concat[Kn * (lane + 32) + t];
        }
        const float inv = 1.0f / (float)Kn;
        Atile[p * A_STRIDE + lane]           = s0 * inv - xc0;
        Atile[p * A_STRIDE + lane + 32]      = s1 * inv - xc1;
        Atile[p * A_STRIDE + 64 + lane]      = xc0;
        Atile[p * A_STRIDE + 64 + lane + 32] = xc1;
    }

    // ---------------- Phase 2: D(16x64) = A(16x128) x W^T(128x64) via V_WMMA_F32_16X16X4_F32 ----
    // f32 A fragment layout (16x4): lanes 0-15 hold row M=lane, K={0,1}; lanes 16-31 hold K={2,3}.
    // f32 B fragment layout (4x16): lanes 0-15 hold col N=lane, K={0,1}; lanes 16-31 hold K={2,3}.
    // So each lane's (K,K+1) pair is a contiguous float2 both in the A-tile row and in a W row.
    v8f acc0 = {0,0,0,0,0,0,0,0};
    v8f acc1 = acc0, acc2 = acc0, acc3 = acc0;

    #pragma unroll 4
    for (int kk = 0; kk < 32; ++kk) {                 // 128 / 4 K-steps
        const float2 av = *(const float2*)&Atile[l15 * A_STRIDE + kk * 4 + 2 * hi];
        const v2f Af = { av.x, av.y };

        const float* wb = W + (size_t)l15 * (2 * Cn) + kk * 4 + 2 * hi;
        const float2 b0 = *(const float2*)(wb + (size_t)0  * 16 * (2 * Cn));
        const float2 b1 = *(const float2*)(wb + (size_t)1  * 16 * (2 * Cn));
        const float2 b2 = *(const float2*)(wb + (size_t)2  * 16 * (2 * Cn));
        const float2 b3 = *(const float2*)(wb + (size_t)3  * 16 * (2 * Cn));
        const v2f B0 = { b0.x, b0.y };
        const v2f B1 = { b1.x, b1.y };
        const v2f B2 = { b2.x, b2.y };
        const v2f B3 = { b3.x, b3.y };

        acc0 = __builtin_amdgcn_wmma_f32_16x16x4_f32(false, Af, false, B0, (short)0, acc0, false, false);
        acc1 = __builtin_amdgcn_wmma_f32_16x16x4_f32(false, Af, false, B1, (short)0, acc1, false, false);
        acc2 = __builtin_amdgcn_wmma_f32_16x16x4_f32(false, Af, false, B2, (short)0, acc2, false, false);
        acc3 = __builtin_amdgcn_wmma_f32_16x16x4_f32(false, Af, false, B3, (short)0, acc3, false, false);
    }

    // ---------------- Phase 3: writeback (+bias) ----------------
    // D layout: lane column N = l15; VGPR v holds rows M = v (lanes 0-15) and M = v+8 (lanes 16-31).
    const float bz0 = bias[ 0 + l15];
    const float bz1 = bias[16 + l15];
    const float bz2 = bias[32 + l15];
    const float bz3 = bias[48 + l15];

    #pragma unroll
    for (int v = 0; v < 8; ++v) {
        const int prow = v + 8 * hi;
        float* o = out + (size_t)(base_point + prow) * On;
        o[ 0 + l15] = acc0[v] + bz0;
        o[16 + l15] = acc1[v] + bz1;
        o[32 + l15] = acc2[v] + bz2;
        o[48 + l15] = acc3[v] + bz3;
    }
}

extern "C" void kernel_launch(void* const* d_in, const int* in_sizes, int n_in,
                              void* d_out, int out_size, void* d_ws, size_t ws_size,
                              hipStream_t stream) {
    const float* x    = (const float*)d_in[0];
    const int*   adj  = (const int*)  d_in[1];
    const float* W    = (const float*)d_in[2];
    const float* bias = (const float*)d_in[3];
    float* out = (float*)d_out;

    const int total_points = Bn * Nn;            // 131072
    const int total_waves  = total_points / 16;  // 8192
    const int blocks       = total_waves / WAVES;// 2048

    EdgeConv_wmma_kernel<<<blocks, WAVES * 32, 0, stream>>>(x, adj, W, bias, out);
}